// BiasedMatrixFactorization_11201274708683
// MI455X (gfx1250) — compile-verified
//
#include <hip/hip_runtime.h>

typedef __attribute__((ext_vector_type(2))) float v2f;
typedef __attribute__((ext_vector_type(8))) float v8f;

#define GLOBAL_AVERAGE 3.5f
#define NF 64  // factors per row

// One wave32 computes 16 batch rows' dot products as diag(Uf16x64 * If64x16)
// via 16 chained V_WMMA_F32_16X16X4_F32 ops.
//
// A (16x4 f32) layout: lanes 0-15 -> M=lane, VGPR0=K0,VGPR1=K1;
//                      lanes 16-31 -> M=lane-16, VGPR0=K2,VGPR1=K3.
// B (4x16 f32) layout mirrors it with N in place of M.
// => every lane loads one contiguous float2 from "its" user row and one from
//    "its" item row per K-chunk: loads land directly in WMMA register layout.
__global__ __launch_bounds__(128) void bmf_wmma_kernel(
    const int*   __restrict__ users,
    const int*   __restrict__ items,
    const float* __restrict__ user_factors,
    const float* __restrict__ item_factors,
    const float* __restrict__ user_biases,
    const float* __restrict__ item_biases,
    float*       __restrict__ out,
    int batch)
{
    const int lane = threadIdx.x & 31;
    const int wave = threadIdx.x >> 5;
    const int row0 = (blockIdx.x * 4 + wave) * 16;   // 16 rows per wave
    if (row0 >= batch) return;                       // wave-uniform branch

    const int mn = lane & 15;          // row-in-group this lane serves (A: M, B: N)
    const int kb = (lane >> 4) * 2;    // K sub-offset: 0 for lanes 0-15, 2 for 16-31

    int r = row0 + mn;
    if (r > batch - 1) r = batch - 1;  // clamp (batch may not be multiple of 16)

    const int urow = users[r];
    const int irow = items[r];

    const float* up = user_factors + (size_t)urow * NF + kb;
    const float* ip = item_factors + (size_t)irow * NF + kb;

    v8f acc = {};
#pragma unroll
    for (int kc = 0; kc < NF; kc += 4) {
        v2f a = *(const v2f*)(up + kc);   // A[m][kc+kb .. kc+kb+1]
        v2f b = *(const v2f*)(ip + kc);   // B[kc+kb .. kc+kb+1][n]
        // 8 args: (neg_a, A, neg_b, B, c_mod, C, reuse_a, reuse_b)
        acc = __builtin_amdgcn_wmma_f32_16x16x4_f32(
            false, a, false, b, (short)0, acc, false, false);
    }

    // Diagonal D[i][i]: i<8 -> lane i, VGPR i ; i>=8 -> lane i+16, VGPR i-8.
    // For all 16 diagonal-holding lanes, i == (lane & 15) == mn.
    const int e = lane & 7;
    float d = acc[0];
    d = (e == 1) ? acc[1] : d;
    d = (e == 2) ? acc[2] : d;
    d = (e == 3) ? acc[3] : d;
    d = (e == 4) ? acc[4] : d;
    d = (e == 5) ? acc[5] : d;
    d = (e == 6) ? acc[6] : d;
    d = (e == 7) ? acc[7] : d;

    const bool diag_lane = (lane < 8) || (lane >= 24);
    if (diag_lane && (row0 + mn) < batch) {
        out[row0 + mn] = GLOBAL_AVERAGE + user_biases[urow] + item_biases[irow] + d;
    }
}

extern "C" void kernel_launch(void* const* d_in, const int* in_sizes, int n_in,
                              void* d_out, int out_size, void* d_ws, size_t ws_size,
                              hipStream_t stream) {
    (void)n_in; (void)out_size; (void)d_ws; (void)ws_size;
    const int*   users = (const int*)  d_in[0];
    const int*   items = (const int*)  d_in[1];
    const float* uf    = (const float*)d_in[2];
    const float* itf   = (const float*)d_in[3];
    const float* ub    = (const float*)d_in[4];
    const float* ib    = (const float*)d_in[5];
    float* out = (float*)d_out;

    const int batch  = in_sizes[0];
    const int waves  = (batch + 15) / 16;   // one wave per 16 rows
    const int blocks = (waves + 3) / 4;     // 4 waves per 128-thread block
    bmf_wmma_kernel<<<blocks, 128, 0, stream>>>(users, items, uf, itf, ub, ib, out, batch);
}